// SWD3_28449863369547
// MI455X (gfx1250) — compile-verified
//
#include <hip/hip_runtime.h>
#include <hip/hip_bf16.h>
#include <stdint.h>

#define SEQ_L 1024
#define DIM_D 64
#define SORT_THREADS 512   // 16 waves (wave32), 2 elements per thread

// ---------------------------------------------------------------------------
// CDNA5 async global -> LDS DMA (gfx1250). Per-lane scattered dword copy,
// tracked with ASYNCcnt. VDST register holds the wave-relative LDS byte
// address, VADDR pair holds the 64-bit global address (SADDR = off/NULL).
// ---------------------------------------------------------------------------
__device__ __forceinline__ void async_gload_dword_to_lds(uint32_t lds_byte_off,
                                                         const float* gptr) {
    asm volatile("global_load_async_to_lds_b32 %0, %1, off"
                 :
                 : "v"(lds_byte_off), "v"((uint64_t)(uintptr_t)gptr)
                 : "memory");
}

__device__ __forceinline__ void wait_async_lds() {
    asm volatile("s_wait_asynccnt 0" ::: "memory");
}

// LDS flat addresses: low 32 bits of the shared-aperture flat address are the
// wave-relative LDS byte offset (ISA 10.2 aperture mapping).
__device__ __forceinline__ uint32_t lds_offset_of(const void* p) {
    return (uint32_t)(uintptr_t)p;
}

// Monotonic bijection float -> u32 (total order matches float <) and inverse.
__device__ __forceinline__ uint32_t f2ord(float f) {
    uint32_t b = __float_as_uint(f);
    return (b & 0x80000000u) ? ~b : (b | 0x80000000u);
}
__device__ __forceinline__ float ord2f(uint32_t u) {
    uint32_t b = (u & 0x80000000u) ? (u ^ 0x80000000u) : ~u;
    return __uint_as_float(b);
}

// ---------------------------------------------------------------------------
// Kernel A: zero the [B,H,L,L] accumulator (output buffer), vectorized.
// ---------------------------------------------------------------------------
__global__ void __launch_bounds__(256) zero_out_kernel(float4* __restrict__ out, int n4) {
    int i = blockIdx.x * blockDim.x + threadIdx.x;
    if (i < n4) {
        out[i] = make_float4(0.f, 0.f, 0.f, 0.f);
    }
}

// ---------------------------------------------------------------------------
// Kernel B: per (bh, d): async-DMA the strided q/k columns into LDS, build
// u64 keys (ordered-float<<32 | index), bitonic-sort both key arrays, then
// scatter-add exp(-(dq-dk)^2)/D at (q_idx[r], k_idx[r]) with fp32 atomics.
// The attention mask is fused here: masked cells must be 0 in the output,
// and the output is zero-initialized, so masked contributions are dropped.
// ---------------------------------------------------------------------------
__global__ void __launch_bounds__(SORT_THREADS)
sort_scatter_kernel(const float* __restrict__ q,
                    const float* __restrict__ k,
                    const unsigned char* __restrict__ mask,
                    float* __restrict__ p) {
    __shared__ float qraw[SEQ_L];
    __shared__ float kraw[SEQ_L];
    __shared__ unsigned long long qkey[SEQ_L];
    __shared__ unsigned long long kkey[SEQ_L];

    const int bh = blockIdx.x;           // 0 .. B*H-1
    const int d  = blockIdx.y;           // 0 .. D-1
    const int t  = threadIdx.x;          // 0 .. 511

    // element (bh, r, d) of [BH, L, D] lives at bh*L*D + r*D + d
    const size_t col_base = (size_t)bh * SEQ_L * DIM_D + (size_t)d;

    const uint32_t qraw_lds = lds_offset_of(qraw);
    const uint32_t kraw_lds = lds_offset_of(kraw);

    // ---- async gather of the two strided columns into LDS -----------------
    #pragma unroll
    for (int s = 0; s < SEQ_L / SORT_THREADS; ++s) {
        const int r = t + s * SORT_THREADS;
        const size_t g = col_base + (size_t)r * DIM_D;
        async_gload_dword_to_lds(qraw_lds + (uint32_t)r * 4u, q + g);
        async_gload_dword_to_lds(kraw_lds + (uint32_t)r * 4u, k + g);
    }
    wait_async_lds();      // this wave's DMA has landed in LDS
    __syncthreads();       // publish to all 16 waves

    // ---- build composite sort keys: ordered(value) << 32 | index ----------
    #pragma unroll
    for (int s = 0; s < SEQ_L / SORT_THREADS; ++s) {
        const int r = t + s * SORT_THREADS;
        qkey[r] = ((unsigned long long)f2ord(qraw[r]) << 32) | (unsigned)r;
        kkey[r] = ((unsigned long long)f2ord(kraw[r]) << 32) | (unsigned)r;
    }
    __syncthreads();

    // ---- bitonic sort (ascending) of both u64 key arrays ------------------
    // 512 threads each own one disjoint compare-exchange pair per step.
    // For stride <= 32, element block [64w, 64w+64) is owned entirely by
    // wave w (wave32), so consecutive small-stride steps need only a
    // wave-local scheduling barrier instead of a workgroup barrier.
    int prev_stride = SEQ_L;   // force full barrier semantics at entry
    for (int size = 2; size <= SEQ_L; size <<= 1) {
        for (int stride = size >> 1; stride > 0; stride >>= 1) {
            if (prev_stride > 32 || stride > 32) {
                __syncthreads();
            } else {
                __builtin_amdgcn_wave_barrier();
            }
            prev_stride = stride;

            const int lo  = (t << 1) - (t & (stride - 1));
            const int hi  = lo + stride;
            const bool asc = ((lo & size) == 0);

            {
                unsigned long long a = qkey[lo], b = qkey[hi];
                if ((a > b) == asc) { qkey[lo] = b; qkey[hi] = a; }
            }
            {
                unsigned long long a = kkey[lo], b = kkey[hi];
                if ((a > b) == asc) { kkey[lo] = b; kkey[hi] = a; }
            }
        }
    }
    __syncthreads();

    // ---- masked scatter ----------------------------------------------------
    // p[bh, qi[r], ki[r]] += exp(-(qv[r]-kv[r])^2)/D   unless mask is set.
    // (masked output cells must be 0; they stay at their zero-init value)
    const size_t p_base = (size_t)bh * SEQ_L * SEQ_L;
    #pragma unroll
    for (int s = 0; s < SEQ_L / SORT_THREADS; ++s) {
        const int r = t + s * SORT_THREADS;
        const unsigned long long a = qkey[r];
        const unsigned long long b = kkey[r];
        const int   qidx = (int)(a & 0xFFFFFFFFull);
        const int   kidx = (int)(b & 0xFFFFFFFFull);
        const size_t cell = p_base + (size_t)qidx * SEQ_L + (size_t)kidx;
        if (!mask[cell]) {
            const float qvf  = ord2f((uint32_t)(a >> 32));
            const float kvf  = ord2f((uint32_t)(b >> 32));
            const float diff = qvf - kvf;
            const float e = __expf(-(diff * diff)) * (1.0f / (float)DIM_D);
            // hardware global_atomic_add_f32, no return, device scope
            __hip_atomic_fetch_add(p + cell, e, __ATOMIC_RELAXED,
                                   __HIP_MEMORY_SCOPE_AGENT);
        }
    }
}

// ---------------------------------------------------------------------------
extern "C" void kernel_launch(void* const* d_in, const int* in_sizes, int n_in,
                              void* d_out, int out_size, void* d_ws, size_t ws_size,
                              hipStream_t stream) {
    const float*         q    = (const float*)d_in[0];          // [B,H,L,D]
    const float*         k    = (const float*)d_in[1];          // [B,H,L,D]
    const unsigned char* mask = (const unsigned char*)d_in[2];  // [B,H,L,L] bool
    float*               out  = (float*)d_out;                  // [B,H,L,L]

    const int BH = in_sizes[0] / (SEQ_L * DIM_D);               // B*H = 64
    const int n4 = out_size / 4;                                // 16M float4

    // 1) zero accumulator (output is poisoned by the harness; masked and
    //    untouched cells must read 0)
    {
        int blocks = (n4 + 255) / 256;
        zero_out_kernel<<<blocks, 256, 0, stream>>>((float4*)out, n4);
    }

    // 2) sort + mask-fused scatter: one block per (bh, d)
    {
        dim3 grid(BH, DIM_D);
        sort_scatter_kernel<<<grid, SORT_THREADS, 0, stream>>>(q, k, mask, out);
    }
}